// GINNet_NC_33200097198350
// MI455X (gfx1250) — compile-verified
//
#include <hip/hip_runtime.h>
#include <math.h>

typedef float v2f __attribute__((ext_vector_type(2)));
typedef float v8f __attribute__((ext_vector_type(8)));

#define DIM 128

// ---------------------------------------------------------------- fill zero
__global__ void fill_zero_f4(float4* __restrict__ p, long long n4) {
    long long i = (long long)blockIdx.x * blockDim.x + threadIdx.x;
    if (i < n4) p[i] = make_float4(0.f, 0.f, 0.f, 0.f);
}

// ---------------------------------------------------------------- edge aggregation
// One wave32 per edge: lane l handles features [4l, 4l+4). 128 = 32 lanes * 4.
__global__ void gin_aggregate(const float* __restrict__ x,
                              const int* __restrict__ src,
                              const int* __restrict__ dst,
                              const float* __restrict__ ew,
                              float* __restrict__ agg,
                              int nEdges, int useW) {
    int e = blockIdx.x * 8 + (threadIdx.x >> 5);
    if (e >= nEdges) return;
    int lane = threadIdx.x & 31;
    int s = src[e];
    int d = dst[e];
    float w = useW ? ew[e] : 1.0f;
    const float4 v = ((const float4*)(x + (size_t)s * DIM))[lane];
    float* ap = agg + (size_t)d * DIM + lane * 4;
#if defined(__HIP_DEVICE_COMPILE__)
    unsafeAtomicAdd(ap + 0, v.x * w);
    unsafeAtomicAdd(ap + 1, v.y * w);
    unsafeAtomicAdd(ap + 2, v.z * w);
    unsafeAtomicAdd(ap + 3, v.w * w);
#else
    atomicAdd(ap + 0, v.x * w);
    atomicAdd(ap + 1, v.y * w);
    atomicAdd(ap + 2, v.z * w);
    atomicAdd(ap + 3, v.w * w);
#endif
}

// ---------------------------------------------------------------- layer GEMM (128 -> 128)
// Block = 256 threads (8 waves). Block handles 16 rows; wave w computes output
// columns [16w, 16w+16) via V_WMMA_F32_16X16X4_F32 over K=128 (32 WMMA ops).
// Fused: h = (1+eps)*x + agg; out = l2norm(relu(h@W + b))  (trailing ReLU is a no-op).
__global__ void gin_gemm128(const float* __restrict__ x,
                            const float* __restrict__ agg,
                            const float* __restrict__ W,
                            const float* __restrict__ bias,
                            const float* __restrict__ epsPtr,
                            float* __restrict__ out, int nNodes) {
    __shared__ float hTile[16][DIM];
    __shared__ float oTile[16][DIM];
    __shared__ float rnorm[16];

    const int tile = blockIdx.x;
    const float eps1 = 1.0f + epsPtr[0];

    // Stage 1: combine into LDS (zero-fill out-of-range rows)
    for (int i = threadIdx.x; i < 16 * DIM; i += 256) {
        int r = i >> 7, c = i & (DIM - 1);
        int g = tile * 16 + r;
        hTile[r][c] = (g < nNodes) ? (eps1 * x[(size_t)g * DIM + c] + agg[(size_t)g * DIM + c]) : 0.0f;
    }
    __syncthreads();

    // Stage 2: WMMA f32 16x16x4, K-loop
    const int wave = threadIdx.x >> 5;
    const int lane = threadIdx.x & 31;
    const int half = lane >> 4;   // 0: K pair {k0,k0+1}; 1: {k0+2,k0+3}
    const int l    = lane & 15;
    const int col0 = wave * 16;

    v8f acc = {};
    for (int k0 = 0; k0 < DIM; k0 += 4) {
        v2f a, b;
        // A 16x4: lanes 0-15 -> M=l, K=k0/k0+1 ; lanes 16-31 -> M=l, K=k0+2/k0+3
        a.x = hTile[l][k0 + 2 * half + 0];
        a.y = hTile[l][k0 + 2 * half + 1];
        // B 4x16: lanes 0-15 -> N=l, K=k0/k0+1 ; lanes 16-31 -> N=l, K=k0+2/k0+3
        b.x = W[(size_t)(k0 + 2 * half + 0) * DIM + col0 + l];
        b.y = W[(size_t)(k0 + 2 * half + 1) * DIM + col0 + l];
        acc = __builtin_amdgcn_wmma_f32_16x16x4_f32(false, a, false, b, (short)0, acc, false, false);
    }

    // Stage 3: bias + ReLU into LDS out tile. C layout: VGPR r -> M = r + 8*half, N = col0+l
    const float bv = bias[col0 + l];
    for (int r = 0; r < 8; ++r) {
        float z = acc[r] + bv;
        oTile[r + 8 * half][col0 + l] = (z > 0.0f) ? z : 0.0f;
    }
    __syncthreads();

    // Stage 4: row L2 norms
    if (threadIdx.x < 16) {
        float s = 0.0f;
        for (int c = 0; c < DIM; ++c) { float v = oTile[threadIdx.x][c]; s += v * v; }
        float n = sqrtf(s);
        rnorm[threadIdx.x] = 1.0f / fmaxf(n, 1e-12f);
    }
    __syncthreads();

    // Stage 5: normalized store (post-norm ReLU is identity: values >= 0, scale > 0)
    for (int i = threadIdx.x; i < 16 * DIM; i += 256) {
        int r = i >> 7, c = i & (DIM - 1);
        int g = tile * 16 + r;
        if (g < nNodes) out[(size_t)g * DIM + c] = oTile[r][c] * rnorm[r];
    }
}

// ---------------------------------------------------------------- final layer (128 -> 16) + softmax
__global__ void gin_gemm_final(const float* __restrict__ x,
                               const float* __restrict__ agg,
                               const float* __restrict__ W,   // [128][16]
                               const float* __restrict__ bias,
                               const float* __restrict__ epsPtr,
                               float* __restrict__ outLogits,
                               float* __restrict__ outProbs, int nNodes) {
    __shared__ float hTile[16][DIM];
    __shared__ float lTile[16][16];
    __shared__ float rmax[16];
    __shared__ float rsum[16];

    const int tile = blockIdx.x;
    const float eps1 = 1.0f + epsPtr[0];

    for (int i = threadIdx.x; i < 16 * DIM; i += 256) {
        int r = i >> 7, c = i & (DIM - 1);
        int g = tile * 16 + r;
        hTile[r][c] = (g < nNodes) ? (eps1 * x[(size_t)g * DIM + c] + agg[(size_t)g * DIM + c]) : 0.0f;
    }
    __syncthreads();

    if (threadIdx.x < 32) {  // wave 0 only: single 16x16 output tile
        const int lane = threadIdx.x;
        const int half = lane >> 4;
        const int l    = lane & 15;
        v8f acc = {};
        for (int k0 = 0; k0 < DIM; k0 += 4) {
            v2f a, b;
            a.x = hTile[l][k0 + 2 * half + 0];
            a.y = hTile[l][k0 + 2 * half + 1];
            b.x = W[(size_t)(k0 + 2 * half + 0) * 16 + l];
            b.y = W[(size_t)(k0 + 2 * half + 1) * 16 + l];
            acc = __builtin_amdgcn_wmma_f32_16x16x4_f32(false, a, false, b, (short)0, acc, false, false);
        }
        const float bv = bias[l];
        for (int r = 0; r < 8; ++r) {
            float z = acc[r] + bv;                       // MLP = Linear + ReLU
            lTile[r + 8 * half][l] = (z > 0.0f) ? z : 0.0f;
        }
    }
    __syncthreads();

    if (threadIdx.x < 16) {
        float m = -3.402823466e+38f;
        for (int c = 0; c < 16; ++c) m = fmaxf(m, lTile[threadIdx.x][c]);
        float s = 0.0f;
        for (int c = 0; c < 16; ++c) s += expf(lTile[threadIdx.x][c] - m);
        rmax[threadIdx.x] = m;
        rsum[threadIdx.x] = s;
    }
    __syncthreads();

    int r = threadIdx.x >> 4, c = threadIdx.x & 15;      // 256 threads = full 16x16 tile
    int g = tile * 16 + r;
    if (g < nNodes) {
        float lg = lTile[r][c];
        outLogits[(size_t)g * 16 + c] = lg;
        outProbs [(size_t)g * 16 + c] = expf(lg - rmax[r]) / rsum[r];
    }
}

// ---------------------------------------------------------------- launch
extern "C" void kernel_launch(void* const* d_in, const int* in_sizes, int n_in,
                              void* d_out, int out_size, void* d_ws, size_t ws_size,
                              hipStream_t stream) {
    const float* x  = (const float*)d_in[0];
    const int*   ei = (const int*)d_in[1];     // (2, E): row 0 = src, row 1 = dst
    const float* ew = (const float*)d_in[2];
    const float* Wp[3] = { (const float*)d_in[3], (const float*)d_in[6], (const float*)d_in[9]  };
    const float* bp[3] = { (const float*)d_in[4], (const float*)d_in[7], (const float*)d_in[10] };
    const float* ep[3] = { (const float*)d_in[5], (const float*)d_in[8], (const float*)d_in[11] };

    const int N = in_sizes[0] / DIM;
    const int E = in_sizes[2];
    const int* src = ei;
    const int* dst = ei + E;

    float* agg = (float*)d_ws;
    float* h0  = agg + (size_t)N * DIM;
    float* h1  = h0  + (size_t)N * DIM;
    float* logits = (float*)d_out;
    float* probs  = logits + (size_t)N * 16;

    const dim3 blk(256);
    const long long n4 = (long long)N * DIM / 4;
    const int zeroBlocks = (int)((n4 + 255) / 256);
    const int aggBlocks  = (E + 7) / 8;
    const int tileBlocks = (N + 15) / 16;

    // Layer 0: aggregate(x, w) -> combine+GEMM+norm -> h0
    fill_zero_f4<<<zeroBlocks, blk, 0, stream>>>((float4*)agg, n4);
    gin_aggregate<<<aggBlocks, blk, 0, stream>>>(x, src, dst, ew, agg, E, 1);
    gin_gemm128<<<tileBlocks, blk, 0, stream>>>(x, agg, Wp[0], bp[0], ep[0], h0, N);

    // Layer 1
    fill_zero_f4<<<zeroBlocks, blk, 0, stream>>>((float4*)agg, n4);
    gin_aggregate<<<aggBlocks, blk, 0, stream>>>(h0, src, dst, ew, agg, E, 1);
    gin_gemm128<<<tileBlocks, blk, 0, stream>>>(h0, agg, Wp[1], bp[1], ep[1], h1, N);

    // Layer 2 (no edge weight) + softmax
    fill_zero_f4<<<zeroBlocks, blk, 0, stream>>>((float4*)agg, n4);
    gin_aggregate<<<aggBlocks, blk, 0, stream>>>(h1, src, dst, ew, agg, E, 0);
    gin_gemm_final<<<tileBlocks, blk, 0, stream>>>(h1, agg, Wp[2], bp[2], ep[2], logits, probs, N);
}